// MAGNN_13391708029877
// MI455X (gfx1250) — compile-verified
//
#include <hip/hip_runtime.h>

// Problem constants (from reference)
#define M_PATHS   6
#define N_INST    20000
#define PATH_LEN  4
#define D         128
#define BLOCKS_PER_M 79          // ceil(20000/256)

// Workspace layout (float offsets)
#define U_OFF 0                          // u[m][d] = W_enc[m] @ Wa  (6*128)
#define V_OFF 768                        // v[m][d] = W_enc[m] @ Wb  (6*128)
#define C_OFF 1536                       // c[m] scalar (pad to 16)
#define S_OFF 1552                       // lrelu(scores)[m][i]      (6*20000)
#define R_OFF (S_OFF + M_PATHS*N_INST)   // per-m {max, 1/sumexp}    (16)
#define G_OFF (R_OFF + 16)               // g[m][d] weighted sum     (6*128)
#define P_OFF (G_OFF + M_PATHS*D)        // per-block partials       (6*79*128)
// total = P_OFF + 6*79*128 = 183008 floats ~= 733 KB of d_ws

typedef __attribute__((ext_vector_type(2))) float v2f;
typedef __attribute__((ext_vector_type(8))) float v8f;

// ---------------------------------------------------------------------------
// K0: per-path fold of W_enc into the attention vectors.
//   u_m = W_enc[m] @ W_att[e,:128], v_m = W_enc[m] @ W_att[e,128:],
//   c_m = b_enc[m]·(Wa+Wb) + b_att[e]
// ---------------------------------------------------------------------------
__global__ void k0_prep(const float* __restrict__ W_enc, const float* __restrict__ b_enc,
                        const float* __restrict__ W_att, const float* __restrict__ b_att,
                        const int* __restrict__ edge_types, float* __restrict__ ws) {
  const int m = blockIdx.x;
  const int d = threadIdx.x;
  const int e = edge_types[m];
  const float* Wa = W_att + e * 2 * D;
  const float* Wb = Wa + D;
  const float* Wrow = W_enc + (m * D + d) * D;   // row d of W_enc[m] (contiguous)
  float u = 0.f, v = 0.f;
  for (int h = 0; h < D; ++h) { float w = Wrow[h]; u += w * Wa[h]; v += w * Wb[h]; }
  ws[U_OFF + m * D + d] = u;
  ws[V_OFF + m * D + d] = v;
  if (d == 0) {
    float c = b_att[e];
    const float* be = b_enc + m * D;
    for (int h = 0; h < D; ++h) c += be[h] * (Wa[h] + Wb[h]);
    ws[C_OFF + m] = c;
  }
}

// ---------------------------------------------------------------------------
// K1: one wave per instance. Gather first/last feature rows (512 B each,
// float4 per lane => one contiguous 512 B wave load), two 128-dots, wave
// reduction, leaky_relu, store score. This pass is the HBM roofline (~123 MB).
// ---------------------------------------------------------------------------
__global__ void k1_scores(const float* __restrict__ features, const int* __restrict__ inst,
                          float* __restrict__ ws) {
  const int wave = threadIdx.x >> 5;
  const int lane = threadIdx.x & 31;
  const int gid  = blockIdx.x * 8 + wave;        // instance id over all paths
  if (gid >= M_PATHS * N_INST) return;
  const int m = gid / N_INST;

  const long long r0 = inst[gid * PATH_LEN + 0];
  const long long r3 = inst[gid * PATH_LEN + 3];
  const float4* f4 = (const float4*)features;
  float4 gf = f4[r0 * (D / 4) + lane];
  float4 gl = f4[r3 * (D / 4) + lane];
  float4 u = ((const float4*)(ws + U_OFF + m * D))[lane];
  float4 v = ((const float4*)(ws + V_OFF + m * D))[lane];

  float p = gf.x * u.x + gf.y * u.y + gf.z * u.z + gf.w * u.w
          + gl.x * v.x + gl.y * v.y + gl.z * v.z + gl.w * v.w;
  for (int off = 16; off > 0; off >>= 1) p += __shfl_xor(p, off, 32);

  if (lane == 0) {
    float s = p + ws[C_OFF + m];
    s = s > 0.f ? s : 0.2f * s;                  // leaky_relu(0.2)
    ws[S_OFF + gid] = s;
  }
}

// ---------------------------------------------------------------------------
// K2: per-path softmax statistics over 20000 scores (max + 1/sum(exp)).
// ---------------------------------------------------------------------------
__global__ void k2_stats(float* __restrict__ ws) {
  const int m = blockIdx.x;
  const int tid = threadIdx.x;
  __shared__ float red[256];
  const float* s = ws + S_OFF + m * N_INST;

  float mx = -3.4e38f;
  for (int i = tid; i < N_INST; i += 256) mx = fmaxf(mx, s[i]);
  red[tid] = mx; __syncthreads();
  for (int k = 128; k > 0; k >>= 1) {
    if (tid < k) red[tid] = fmaxf(red[tid], red[tid + k]);
    __syncthreads();
  }
  mx = red[0]; __syncthreads();

  float sum = 0.f;
  for (int i = tid; i < N_INST; i += 256) sum += expf(s[i] - mx);
  red[tid] = sum; __syncthreads();
  for (int k = 128; k > 0; k >>= 1) {
    if (tid < k) red[tid] += red[tid + k];
    __syncthreads();
  }
  if (tid == 0) { ws[R_OFF + m * 2 + 0] = mx; ws[R_OFF + m * 2 + 1] = 1.f / red[0]; }
}

// ---------------------------------------------------------------------------
// K3: weighted feature sum g_m = sum_i w_i * gl_i. One block per 256
// instances of one path; 8 waves each stream 32 instances (re-gather hits L2,
// footprint <= 123 MB < 192 MB). Deterministic: fixed-order LDS tree, then
// per-block partial to workspace (no atomics).
// ---------------------------------------------------------------------------
__global__ void k3_wsum(const float* __restrict__ features, const int* __restrict__ inst,
                        float* __restrict__ ws) {
  const int m     = blockIdx.x / BLOCKS_PER_M;
  const int chunk = blockIdx.x % BLOCKS_PER_M;
  const int wave  = threadIdx.x >> 5;
  const int lane  = threadIdx.x & 31;
  const int base  = chunk * 256 + wave * 32;

  const float mx  = ws[R_OFF + m * 2 + 0];
  const float inv = ws[R_OFF + m * 2 + 1];
  const float4* f4 = (const float4*)features;

  float4 acc = {0.f, 0.f, 0.f, 0.f};
  for (int j = 0; j < 32; ++j) {
    const int i = base + j;
    if (i >= N_INST) break;                       // wave-uniform
    const int gid = m * N_INST + i;
    const float wgt = expf(ws[S_OFF + gid] - mx) * inv;
    const long long r3 = inst[gid * PATH_LEN + 3];
    float4 gl = f4[r3 * (D / 4) + lane];
    acc.x += wgt * gl.x; acc.y += wgt * gl.y; acc.z += wgt * gl.z; acc.w += wgt * gl.w;
  }

  __shared__ float sh[8][D];
  sh[wave][lane * 4 + 0] = acc.x;
  sh[wave][lane * 4 + 1] = acc.y;
  sh[wave][lane * 4 + 2] = acc.z;
  sh[wave][lane * 4 + 3] = acc.w;
  __syncthreads();
  if (threadIdx.x < D) {
    float t = 0.f;
    for (int w = 0; w < 8; ++w) t += sh[w][threadIdx.x];   // fixed order
    ws[P_OFF + blockIdx.x * D + threadIdx.x] = t;
  }
}

// K3b: fold per-block partials into g_m (fixed order => deterministic).
__global__ void k3b_reduce(float* __restrict__ ws) {
  const int m = blockIdx.x, h = threadIdx.x;
  float t = 0.f;
  for (int b = 0; b < BLOCKS_PER_M; ++b)
    t += ws[P_OFF + (m * BLOCKS_PER_M + b) * D + h];
  ws[G_OFF + m * D + h] = t;
}

// ---------------------------------------------------------------------------
// K4: the one surviving dense matrix op, done with V_WMMA_F32_16X16X4_F32:
//   mp_out[m] = g_m @ W_enc[m] + b_enc[m]   (M padded to 16, row 0 live)
// 8 waves x 16-column tiles, K-loop of 32 wmma. Then path softmax + ELU.
// A layout per ISA: lanes0-15 {K0,K1}, lanes16-31 {K2,K3}, M = lane&15.
// B layout assumed mirrored: lane half selects K pair, VGPR0/1 = consecutive K rows.
// g is staged in LDS so the A fragment is built from unconditional ds_load +
// v_cndmask (no exec-divergent branches; EXEC stays all-ones around wmma).
// ---------------------------------------------------------------------------
__global__ void k4_final(const float* __restrict__ W_enc, const float* __restrict__ b_enc,
                         const float* __restrict__ w_mp, const float* __restrict__ b_mp,
                         const float* __restrict__ ws, float* __restrict__ out) {
  __shared__ float gall[M_PATHS][D];   // weighted feature sums, staged once
  __shared__ float mp[M_PATHS][D];
  __shared__ float msc[8];
  const int wave = threadIdx.x >> 5;
  const int lane = threadIdx.x & 31;
  const int half = lane >> 4;
  const int col  = lane & 15;     // A's M index and B/C's N index
  const int n0   = wave * 16;     // this wave's column tile
  const bool row0 = (col == 0);   // only live row of the padded A tile

  for (int t = threadIdx.x; t < M_PATHS * D; t += 256)
    gall[t / D][t % D] = ws[G_OFF + t];
  __syncthreads();

  for (int m = 0; m < M_PATHS; ++m) {
    const float* W = W_enc + m * D * D;
    v8f c = {0.f, 0.f, 0.f, 0.f, 0.f, 0.f, 0.f, 0.f};
#pragma unroll 4
    for (int k = 0; k < D; k += 4) {
      // unconditional 8B LDS load of {g[k+2h], g[k+2h+1]}, then cndmask-select
      const float2 gp = *(const float2*)&gall[m][k + 2 * half];
      v2f a;
      a.x = row0 ? gp.x : 0.f;
      a.y = row0 ? gp.y : 0.f;
      v2f b;                                       // B 4x16 tile of W_enc[m]
      b.x = W[(k + 2 * half + 0) * D + n0 + col];
      b.y = W[(k + 2 * half + 1) * D + n0 + col];
      c = __builtin_amdgcn_wmma_f32_16x16x4_f32(false, a, false, b, (short)0, c,
                                                false, false);
    }
    // C/D layout: VGPR0, lanes 0..15 hold M=0, N=lane
    if (lane < 16) mp[m][n0 + lane] = c[0] + b_enc[m * D + n0 + lane];
  }
  __syncthreads();

  if (threadIdx.x < M_PATHS) {
    float s = b_mp[0];
    for (int h = 0; h < D; ++h) s += mp[threadIdx.x][h] * w_mp[h];
    msc[threadIdx.x] = s > 0.f ? s : 0.2f * s;     // leaky_relu
  }
  __syncthreads();
  if (threadIdx.x == 0) {
    float mx = msc[0];
    for (int m = 1; m < M_PATHS; ++m) mx = fmaxf(mx, msc[m]);
    float sum = 0.f;
    for (int m = 0; m < M_PATHS; ++m) { msc[m] = expf(msc[m] - mx); sum += msc[m]; }
    for (int m = 0; m < M_PATHS; ++m) msc[m] /= sum;
  }
  __syncthreads();
  if (threadIdx.x < D) {
    float o = 0.f;
    for (int m = 0; m < M_PATHS; ++m) o += msc[m] * mp[m][threadIdx.x];
    out[threadIdx.x] = o > 0.f ? o : (expf(o) - 1.f);  // ELU
  }
}

// ---------------------------------------------------------------------------
extern "C" void kernel_launch(void* const* d_in, const int* in_sizes, int n_in,
                              void* d_out, int out_size, void* d_ws, size_t ws_size,
                              hipStream_t stream) {
  const float* features  = (const float*)d_in[0];
  const float* W_enc     = (const float*)d_in[1];
  const float* b_enc     = (const float*)d_in[2];
  const float* W_att     = (const float*)d_in[3];
  const float* b_att     = (const float*)d_in[4];
  const float* w_mp      = (const float*)d_in[5];
  const float* b_mp      = (const float*)d_in[6];
  const int*   instances = (const int*)d_in[7];
  const int*   edge_types= (const int*)d_in[8];
  float* ws  = (float*)d_ws;
  float* out = (float*)d_out;

  k0_prep  <<<M_PATHS, D,   0, stream>>>(W_enc, b_enc, W_att, b_att, edge_types, ws);
  k1_scores<<<(M_PATHS * N_INST + 7) / 8, 256, 0, stream>>>(features, instances, ws);
  k2_stats <<<M_PATHS, 256, 0, stream>>>(ws);
  k3_wsum  <<<M_PATHS * BLOCKS_PER_M, 256, 0, stream>>>(features, instances, ws);
  k3b_reduce<<<M_PATHS, D,  0, stream>>>(ws);
  k4_final <<<1, 256,       0, stream>>>(W_enc, b_enc, w_mp, b_mp, ws, out);
}